// VoronoiWatershedLoss_64982855189206
// MI455X (gfx1250) — compile-verified
//
#include <hip/hip_runtime.h>
#include <math.h>

#define EPSF 1e-7f

typedef __attribute__((ext_vector_type(2))) float v2f;
typedef __attribute__((ext_vector_type(8))) float v8f;

// ---------------------------------------------------------------------------
// Kernel A: per-object preprocessing (eig, normalized sigma, inverse, poly
// coefficients for the expanded quadratic form) + zero-init of reduction slots
// ---------------------------------------------------------------------------
__global__ __launch_bounds__(256) void vwl_prep(
    const float* __restrict__ mu, const float* __restrict__ sigma,
    float* __restrict__ coef, float* __restrict__ Vout, float* __restrict__ Lout,
    int* __restrict__ gmx, int* __restrict__ gflag)
{
  const int t = threadIdx.x;
  if (t < 128) gmx[t] = 0;
  if (t < 64)  gflag[t] = 0;
  if (t < 64) {
    const int j = t;
    float a = sigma[j * 4 + 0];
    float b = sigma[j * 4 + 1];   // symmetric: == sigma[j*4+2]
    float c = sigma[j * 4 + 3];
    // 2x2 symmetric eigendecomposition, ascending (matches jnp.linalg.eigh)
    float ht = 0.5f * (a + c);
    float df = 0.5f * (a - c);
    float disc = sqrtf(df * df + b * b);
    float l0 = ht - disc, l1 = ht + disc;
    float v1x, v1y;
    if (fabsf(b) > 1e-30f) { v1x = b; v1y = l1 - a; }
    else { v1x = (a >= c) ? 1.0f : 0.0f; v1y = (a >= c) ? 0.0f : 1.0f; }
    float nrm = sqrtf(v1x * v1x + v1y * v1y);
    v1x /= nrm; v1y /= nrm;
    float v0x = -v1y, v0y = v1x;               // eigvec for l0 (orthogonal)
    // V row-major: V[k][i], column i = eigenvector i
    Vout[j * 4 + 0] = v0x; Vout[j * 4 + 1] = v1x;
    Vout[j * 4 + 2] = v0y; Vout[j * 4 + 3] = v1y;
    Lout[j * 2 + 0] = l0;  Lout[j * 2 + 1] = l1;
    // Ln = L / sqrt(L0*L1) * 4096 ; sg = V diag(Ln) V^T / 4
    float gm = sqrtf(l0 * l1);
    float Ln0 = l0 / gm * 4096.0f, Ln1 = l1 / gm * 4096.0f;
    float sg00 = (v0x * v0x * Ln0 + v1x * v1x * Ln1) * 0.25f;
    float sg01 = (v0x * v0y * Ln0 + v1x * v1y * Ln1) * 0.25f;
    float sg11 = (v0y * v0y * Ln0 + v1y * v1y * Ln1) * 0.25f;
    float det = sg00 * sg11 - sg01 * sg01;
    float i00 =  sg11 / det, i01 = -sg01 / det, i11 = sg00 / det;
    float mx = rintf(mu[j * 2 + 0] * 0.5f);
    float my = rintf(mu[j * 2 + 1] * 0.5f);
    // q = A x^2 + B xy + C y^2 + D x + E y + F
    float A = i00, Bc = 2.0f * i01, C = i11;
    coef[j * 8 + 0] = A;
    coef[j * 8 + 1] = Bc;
    coef[j * 8 + 2] = C;
    coef[j * 8 + 3] = -2.0f * A * mx - Bc * my;
    coef[j * 8 + 4] = -Bc * mx - 2.0f * C * my;
    coef[j * 8 + 5] = A * mx * mx + Bc * mx * my + C * my * my;
    coef[j * 8 + 6] = 0.0f;
    coef[j * 8 + 7] = 0.0f;
  }
}

// ---------------------------------------------------------------------------
// Kernel B: Gaussian Voronoi render at 512x512 via FP32 WMMA.
// q[16 pix][64 j] = [16x8 poly terms] x [8x64 coeffs] -> two 16x16x4 WMMAs
// per j-tile. Coefficients staged in LDS; B operands built with address
// arithmetic (no divergent loads -> EXEC stays all-ones around the WMMAs).
// ---------------------------------------------------------------------------
__global__ __launch_bounds__(256) void vwl_render(
    const float* __restrict__ coef,
    unsigned char* __restrict__ vor, float* __restrict__ val)
{
  __shared__ float scoef[512];      // 64 objects x 8 coeffs
  __shared__ float qs[8][16][65];   // pad 64->65 to dodge LDS bank conflicts
  const int t = threadIdx.x;
  scoef[t]       = coef[t];
  scoef[t + 256] = coef[t + 256];
  __syncthreads();

  const int wave = t >> 5;
  const int lane = t & 31;
  const int mtile = blockIdx.x * 8 + wave;
  const int m = lane & 15;
  const bool hi = lane >= 16;
  const int pm = mtile * 16 + m;            // pixel id (row-major 512x512)
  const int row = pm >> 9;
  const int col = pm & 511;
  const float s = 512.0f / 511.0f;          // linspace(0, 512, 512) step
  const float x = col * s;
  const float y = row * s;

  // A 16x4 f32 layout: lanes0-15 -> VGPR0=K0,VGPR1=K1 ; lanes16-31 -> K2,K3
  v2f a0, a1;
  a0.x = hi ? y * y : x * x;   // K2 : K0
  a0.y = hi ? x     : x * y;   // K3 : K1
  a1.x = hi ? 0.0f  : y;       // K6 : K4
  a1.y = hi ? 0.0f  : 1.0f;    // K7 : K5

  const int off = hi ? 2 : 0;  // lane-group selects K-pair via address math
#pragma unroll
  for (int nt = 0; nt < 4; ++nt) {
    const int jb = (nt * 16 + m) * 8;
    v2f b0, b1;
    b0.x = scoef[jb + off];        // K0/K2 row of B
    b0.y = scoef[jb + off + 1];    // K1/K3 row of B
    const float c4 = scoef[jb + 4];
    const float c5 = scoef[jb + 5];
    b1.x = hi ? 0.0f : c4;         // K6 : K4
    b1.y = hi ? 0.0f : c5;         // K7 : K5
    v8f acc = {0.f, 0.f, 0.f, 0.f, 0.f, 0.f, 0.f, 0.f};
    acc = __builtin_amdgcn_wmma_f32_16x16x4_f32(false, a0, false, b0,
                                                (short)0, acc, false, false);
    acc = __builtin_amdgcn_wmma_f32_16x16x4_f32(false, a1, false, b1,
                                                (short)0, acc, false, false);
    // C layout: VGPR r -> M = r + (hi?8:0), N = lane&15
    const int mb = hi ? 8 : 0;
#pragma unroll
    for (int r = 0; r < 8; ++r)
      qs[wave][mb + r][nt * 16 + m] = acc[r];
  }
  __syncthreads();

  if (lane < 16) {
    // argmax_j exp(-q/2) == argmin_j q (first-occurrence tie break)
    float best = qs[wave][lane][0];
    int bj = 0;
    for (int jj = 1; jj < 64; ++jj) {
      float q = qs[wave][lane][jj];
      if (q < best) { best = q; bj = jj; }
    }
    vor[pm] = (unsigned char)bj;
    val[pm] = expf(-0.5f * best);
  }
}

// ---------------------------------------------------------------------------
// Kernel C: full-res (1024x1024) upsample + ridge + markers; writes markers to
// d_out[1..] as float and feeds the per-object rotated-extent segment max.
// ---------------------------------------------------------------------------
__global__ __launch_bounds__(256) void vwl_markers(
    const unsigned char* __restrict__ vor, const float* __restrict__ val,
    const int* __restrict__ label, const float* __restrict__ pos_thres,
    const float* __restrict__ neg_thres, const float* __restrict__ mu,
    const float* __restrict__ V, float* __restrict__ out_markers,
    int* __restrict__ gmx, int* __restrict__ gflag)
{
  __shared__ int lmx[128];
  __shared__ int lflag[64];
  const int t = threadIdx.x;
  if (t < 128) lmx[t] = 0;
  if (t < 64)  lflag[t] = 0;
  __syncthreads();

  const int p  = blockIdx.x * 256 + t;
  const int yi = p >> 10;
  const int xi = p & 1023;

  const int vc = vor[(yi >> 1) * 512 + (xi >> 1)];   // nearest 2x upsample

  // bilinear align_corners upsample of val: src = idx * 511/1023
  const float ysf = yi * (511.0f / 1023.0f);
  const float xsf = xi * (511.0f / 1023.0f);
  const int y0 = (int)floorf(ysf), x0 = (int)floorf(xsf);
  const int y1 = min(y0 + 1, 511), x1 = min(x0 + 1, 511);
  const float wy = ysf - (float)y0, wx = xsf - (float)x0;
  const float v00 = val[y0 * 512 + x0], v01 = val[y0 * 512 + x1];
  const float v10 = val[y1 * 512 + x0], v11 = val[y1 * 512 + x1];
  const float vv = v00 * (1.f - wy) * (1.f - wx) + v01 * (1.f - wy) * wx
                 + v10 * wy * (1.f - wx)         + v11 * wy * wx;

  // 3x3 Laplacian ridge on full-res vor (zero padding), exact in integers
  int lap = -8 * vc;
#pragma unroll
  for (int dy = -1; dy <= 1; ++dy)
#pragma unroll
    for (int dx = -1; dx <= 1; ++dx) {
      if (dx == 0 && dy == 0) continue;
      const int yy = yi + dy, xx = xi + dx;
      int nb = 0;
      if (yy >= 0 && yy < 1024 && xx >= 0 && xx < 1024)
        nb = vor[(yy >> 1) * 512 + (xx >> 1)];
      lap += nb;
    }
  const bool ridge = (lap != 0);

  const int cls = label[vc];
  int mk = vc + 1;
  if (vv < pos_thres[cls]) mk = 0;
  if (vv < neg_thres[cls]) mk = 65;   // Jn + 1
  if (ridge)               mk = 65;
  out_markers[p] = (float)mk;

  if (mk >= 1 && mk <= 64) {
    const int j = mk - 1;
    const float dd0 = (float)xi - mu[j * 2 + 0];
    const float dd1 = (float)yi - mu[j * 2 + 1];
    // r_i = | sum_k V[k][i] * dd_k |   (non-negative -> int-bit max is valid)
    const float r0 = fabsf(V[j * 4 + 0] * dd0 + V[j * 4 + 2] * dd1);
    const float r1 = fabsf(V[j * 4 + 1] * dd0 + V[j * 4 + 3] * dd1);
    atomicMax(&lmx[j * 2 + 0], __float_as_int(r0));
    atomicMax(&lmx[j * 2 + 1], __float_as_int(r1));
    lflag[j] = 1;
  }
  __syncthreads();
  if (t < 128 && lflag[t >> 1]) atomicMax(&gmx[t], lmx[t]);
  if (t < 64 && lflag[t]) gflag[t] = 1;
}

// ---------------------------------------------------------------------------
// Kernel D: GWD sigma loss over 64 objects, sort, mean of smallest 61
// ---------------------------------------------------------------------------
__global__ __launch_bounds__(64) void vwl_loss(
    const float* __restrict__ Lin, const int* __restrict__ gmx,
    const int* __restrict__ gflag, float* __restrict__ out)
{
  __shared__ float lv[64];
  const int j = threadIdx.x;
  const float L0 = Lin[j * 2 + 0], L1 = Lin[j * 2 + 1];
  float Lt0, Lt1;
  if (gflag[j]) {
    const float m0 = __int_as_float(gmx[j * 2 + 0]);
    const float m1 = __int_as_float(gmx[j * 2 + 1]);
    Lt0 = m0 * m0; Lt1 = m1 * m1;
  } else {
    Lt0 = L0; Lt1 = L1;
  }
  const float tr = L0 + L1 + Lt0 + Lt1;
  const float t_tr = L0 * Lt0 + L1 * Lt1;
  const float det_sqrt = sqrtf(fmaxf(L0 * L1 * Lt0 * Lt1, EPSF));
  const float whr = tr - 2.0f * sqrtf(fmaxf(t_tr + 2.0f * det_sqrt, EPSF));
  float dist = sqrtf(fmaxf(whr, EPSF));
  const float s1 = sqrtf(fmaxf(det_sqrt, EPSF));
  const float scale = 2.0f * fmaxf(sqrtf(fmaxf(s1, EPSF)), EPSF);
  dist /= scale;
  lv[j] = 1.0f - 1.0f / (1.0f + log1pf(dist));
  __syncthreads();
  if (j == 0) {
    for (int i = 1; i < 64; ++i) {          // insertion sort (ascending)
      float key = lv[i];
      int k = i - 1;
      while (k >= 0 && lv[k] > key) { lv[k + 1] = lv[k]; --k; }
      lv[k + 1] = key;
    }
    float sum = 0.0f;                        // k = ceil(64*0.95) = 61
    for (int i = 0; i < 61; ++i) sum += lv[i];
    out[0] = sum / 61.0f;
  }
}

extern "C" void kernel_launch(void* const* d_in, const int* in_sizes, int n_in,
                              void* d_out, int out_size, void* d_ws, size_t ws_size,
                              hipStream_t stream) {
  const float* mu        = (const float*)d_in[0];   // [64,2]
  const float* sigma     = (const float*)d_in[1];   // [64,2,2]
  const int*   label     = (const int*)d_in[2];     // [64]
  /* d_in[3] = image: unused (feeds skipped cv2.watershed only) */
  const float* pos_thres = (const float*)d_in[4];   // [16]
  const float* neg_thres = (const float*)d_in[5];   // [16]
  float* out = (float*)d_out;                       // [0]=loss, [1..]=markers

  char* ws = (char*)d_ws;
  float*         coef  = (float*)(ws + 0);            // 64*8 f32
  float*         V     = (float*)(ws + 2048);         // 64*4 f32
  float*         L     = (float*)(ws + 3072);         // 64*2 f32
  int*           gmx   = (int*)(ws + 3584);           // 128 i32 (float bits)
  int*           gflag = (int*)(ws + 4096);           // 64 i32
  unsigned char* vor   = (unsigned char*)(ws + 8192); // 512*512 u8
  float*         val   = (float*)(ws + 8192 + 262144);// 512*512 f32

  vwl_prep<<<1, 256, 0, stream>>>(mu, sigma, coef, V, L, gmx, gflag);
  vwl_render<<<2048, 256, 0, stream>>>(coef, vor, val);        // 16384 M-tiles
  vwl_markers<<<4096, 256, 0, stream>>>(vor, val, label, pos_thres, neg_thres,
                                        mu, V, out + 1, gmx, gflag);
  vwl_loss<<<1, 64, 0, stream>>>(L, gmx, gflag, out);
}